// LocalDotAttention_5755256177188
// MI455X (gfx1250) — compile-verified
//
#include <hip/hip_runtime.h>
#include <math.h>

// ---------------------------------------------------------------------------
// LocalDotAttention on MI455X (gfx1250, wave32).
// B=32, TX=4096, D=1024, WINDOW=4 (8 gathered positions).
// All GEMMs use V_WMMA_F32_16X16X4_F32 (fp32 in/out, matches reference dtype).
// ---------------------------------------------------------------------------

typedef __attribute__((ext_vector_type(2))) float v2f;
typedef __attribute__((ext_vector_type(4))) float v4f;
typedef __attribute__((ext_vector_type(8))) float v8f;

#define BQ 32
#define TXQ 4096
#define DQ 1024
#define NW 8   // 2*WINDOW

// K-loop for one 16x16 fp32 tile: C += A[16xK] * W^T tile.
// A: row-major [M x lda] activations.
// W: row-major [N x ldw] weights used transposed -> B[k][n] = W[n0+n][k].
// fp32 WMMA A layout: lanes 0-15 hold M=lane,K={k,k+1}; lanes 16-31 hold
// M=lane-16,K={k+2,k+3}. B layout mirrors it with N instead of M. Both are
// contiguous float2 loads per lane from row-major storage.
__device__ __forceinline__ v8f wmma_k_loop(const float* __restrict__ A, int lda,
                                           const float* __restrict__ W, int ldw,
                                           int m0, int n0, int K, v8f acc) {
  const int lane  = threadIdx.x & 31;
  const int half2 = (lane >> 4) << 1;   // 0 or 2
  const int l16   = lane & 15;
  const float* aptr = A + (size_t)(m0 + l16) * lda + half2;
  const float* wptr = W + (size_t)(n0 + l16) * ldw + half2;
#pragma unroll 8
  for (int k = 0; k < K; k += 4) {
    v2f a = *(const v2f*)(aptr + k);
    v2f b = *(const v2f*)(wptr + k);
    acc = __builtin_amdgcn_wmma_f32_16x16x4_f32(
        /*neg_a=*/false, a, /*neg_b=*/false, b,
        /*c_mod=*/(short)0, acc, /*reuse_a=*/false, /*reuse_b=*/false);
  }
  return acc;
}

// C/D fp32 16x16 layout: VGPR r -> lanes 0-15: (M=r, N=lane),
// lanes 16-31: (M=r+8, N=lane-16).
__device__ __forceinline__ void store_tile(float* __restrict__ C, int ldc,
                                           int m0, int n0, v8f acc,
                                           bool apply_tanh) {
  const int lane = threadIdx.x & 31;
  const int mofs = (lane >> 4) << 3;    // 0 or 8
  const int l16  = lane & 15;
#pragma unroll
  for (int r = 0; r < 8; ++r) {
    float v = acc[r];
    if (apply_tanh) v = tanhf(v);
    C[(size_t)(m0 + r + mofs) * ldc + n0 + l16] = v;
  }
}

// Generic small GEMM: C[M=32 x N] = act(A[32 x K] @ W[N x K]^T).
// One wave per 16x16 tile; wid bit0 = M tile, rest = N tile.
__global__ void ldattn_wmma_gemm(const float* __restrict__ A, int lda,
                                 const float* __restrict__ W, int ldw,
                                 float* __restrict__ C, int ldc,
                                 int K, int apply_tanh) {
  const int wid = (int)((blockIdx.x * blockDim.x + threadIdx.x) >> 5);
  const int m0 = (wid & 1) * 16;
  const int n0 = (wid >> 1) * 16;
  v8f acc = {};
  acc = wmma_k_loop(A, lda, W, ldw, m0, n0, K, acc);
  store_tile(C, ldc, m0, n0, acc, apply_tanh != 0);
}

// pt = TX * sigmoid(p @ W_p2^T); bl = trunc(pt - 4); idx = clip(bl+w, 0, TX-1)
__global__ void ldattn_pt_idx(const float* __restrict__ p,      // [32 x 512]
                              const float* __restrict__ W_p2,   // [512]
                              int* __restrict__ idx) {          // [32 x 8]
  __shared__ float part[256];
  const int tid = threadIdx.x;          // 256 threads, 1 block
  const int b = tid >> 3;               // 8 threads per batch row
  const int c = tid & 7;                // each sums 64 elements
  float s = 0.0f;
  const float* pr = p + b * 512 + c * 64;
  const float* wr = W_p2 + c * 64;
#pragma unroll 8
  for (int j = 0; j < 64; ++j) s += pr[j] * wr[j];
  part[tid] = s;
  __syncthreads();
  if (c == 0) {
    float tot = 0.0f;
#pragma unroll
    for (int i = 0; i < 8; ++i) tot += part[(b << 3) + i];
    float pt = (float)TXQ / (1.0f + expf(-tot));
    int bl = (int)truncf(pt - 4.0f);    // trunc toward zero, like jnp.trunc+int32
#pragma unroll
    for (int w = 0; w < NW; ++w) {
      int v = bl + w;
      v = v < 0 ? 0 : (v > TXQ - 1 ? TXQ - 1 : v);
      idx[b * NW + w] = v;
    }
  }
}

// Gather 8 context rows, scores vs target, softmax, weighted sum.
// One block (256 threads) per batch row; each thread owns 4 columns.
__global__ void ldattn_attend(const float* __restrict__ context, // [32,4096,1024]
                              const float* __restrict__ target,  // [32,1024]
                              const int* __restrict__ idx,       // [32,8]
                              float* __restrict__ weighted,      // [32,1024]
                              float* __restrict__ attn_out) {    // [32,8]
  const int b = blockIdx.x;
  const int tid = threadIdx.x;
  __shared__ float red[256];
  __shared__ float sc[NW];
  __shared__ float aw[NW];

  const int d0 = tid * 4;
  v4f t = *(const v4f*)(target + (size_t)b * DQ + d0);

  v4f cw[NW];
  float partial[NW];
#pragma unroll
  for (int w = 0; w < NW; ++w) {
    const size_t row = ((size_t)b * TXQ + (size_t)idx[b * NW + w]) * DQ;
    v4f c = *(const v4f*)(context + row + d0);
    cw[w] = c;
    partial[w] = c.x * t.x + c.y * t.y + c.z * t.z + c.w * t.w;
  }

  // Block reductions (one per window position).
  for (int w = 0; w < NW; ++w) {
    red[tid] = partial[w];
    __syncthreads();
    for (int s2 = 128; s2 > 0; s2 >>= 1) {
      if (tid < s2) red[tid] += red[tid + s2];
      __syncthreads();
    }
    if (tid == 0) sc[w] = red[0];
    __syncthreads();
  }

  if (tid == 0) {
    float m = sc[0];
#pragma unroll
    for (int w = 1; w < NW; ++w) m = fmaxf(m, sc[w]);
    float e[NW], sum = 0.0f;
#pragma unroll
    for (int w = 0; w < NW; ++w) { e[w] = expf(sc[w] - m); sum += e[w]; }
    float inv = 1.0f / sum;
#pragma unroll
    for (int w = 0; w < NW; ++w) {
      aw[w] = e[w] * inv;
      attn_out[b * NW + w] = aw[w];
    }
  }
  __syncthreads();

  v4f acc = {};
#pragma unroll
  for (int w = 0; w < NW; ++w) {
    float a = aw[w];
    acc.x += a * cw[w].x; acc.y += a * cw[w].y;
    acc.z += a * cw[w].z; acc.w += a * cw[w].w;
  }
  *(v4f*)(weighted + (size_t)b * DQ + d0) = acc;
}

// h_tilde = tanh(concat([weighted, input]) @ W_out^T). K=2048 split into two
// 1024 passes over the same accumulator (second half uses W_out column 1024+).
__global__ void ldattn_out_gemm(const float* __restrict__ weighted,
                                const float* __restrict__ input,
                                const float* __restrict__ W_out, // [1024 x 2048]
                                float* __restrict__ h) {         // [32 x 1024]
  const int wid = (int)((blockIdx.x * blockDim.x + threadIdx.x) >> 5);
  const int m0 = (wid & 1) * 16;
  const int n0 = (wid >> 1) * 16;
  v8f acc = {};
  acc = wmma_k_loop(weighted, DQ, W_out,        2 * DQ, m0, n0, DQ, acc);
  acc = wmma_k_loop(input,    DQ, W_out + DQ,   2 * DQ, m0, n0, DQ, acc);
  store_tile(h, DQ, m0, n0, acc, /*tanh=*/true);
}

extern "C" void kernel_launch(void* const* d_in, const int* in_sizes, int n_in,
                              void* d_out, int out_size, void* d_ws, size_t ws_size,
                              hipStream_t stream) {
  const float* input   = (const float*)d_in[0]; // [32,1024]
  const float* context = (const float*)d_in[1]; // [32,4096,1024]
  const float* W_in    = (const float*)d_in[2]; // [1024,1024]
  const float* W_out   = (const float*)d_in[3]; // [1024,2048]
  const float* W_p1    = (const float*)d_in[4]; // [512,1024]
  const float* W_p2    = (const float*)d_in[5]; // [512]

  float* h_tilde = (float*)d_out;               // [32,1024]
  float* attn    = h_tilde + BQ * DQ;           // [32,8]

  // Workspace layout (fp32): p[32x512] | target[32x1024] | weighted[32x1024] | idx[32x8]
  float* p        = (float*)d_ws;
  float* target   = p + BQ * (DQ / 2);
  float* weighted = target + BQ * DQ;
  int*   idx      = (int*)(weighted + BQ * DQ);

  // 1) p = tanh(input @ W_p1.T)  M=32,N=512,K=1024 -> 64 waves
  ldattn_wmma_gemm<<<16, 128, 0, stream>>>(input, DQ, W_p1, DQ, p, DQ / 2, DQ, 1);
  // 2) predictive position -> window indices
  ldattn_pt_idx<<<1, 256, 0, stream>>>(p, W_p2, idx);
  // 3) target = input @ W_in.T  M=32,N=1024,K=1024 -> 128 waves
  ldattn_wmma_gemm<<<32, 128, 0, stream>>>(input, DQ, W_in, DQ, target, DQ, DQ, 0);
  // 4) gather + softmax + weighted context
  ldattn_attend<<<32, 256, 0, stream>>>(context, target, idx, weighted, attn);
  // 5) h_tilde = tanh([weighted, input] @ W_out.T)  K=2048 -> 128 waves
  ldattn_out_gemm<<<32, 128, 0, stream>>>(weighted, input, W_out, h_tilde);
}